// MPCLayer_51685636440507
// MI455X (gfx1250) — compile-verified
//
#include <hip/hip_runtime.h>

// ---------------------------------------------------------------------------
// MPC / box-constrained LQR, B=65536 independent problems, NS=NA=4, T=5.
//
// Key structure: C is diagonal and Fm is constant => the Riccati quadratic
// recursion (V_t, Quu_t, K_t) is identical for every batch element and every
// LQR iteration. We compute it ONCE on one wave using the fp32 WMMA pipe
// (F' V F is an 8x8,K=4 GEMM -> V_WMMA_F32_16X16X4_F32), store the constant
// gains in d_ws, and run the per-batch linear/clipped sweeps one-thread-per-
// problem, fully register resident.
// ---------------------------------------------------------------------------

typedef __attribute__((ext_vector_type(2))) float v2f;
typedef __attribute__((ext_vector_type(8))) float v8f;

#define T_HORIZON 5

// F = [Ad | Bd], Ad = I + A1*DT, Bd = B1*DT*SENS  (DT=0.1, TAU/DT=2.5, SENS=5)
__device__ __constant__ float c_F[4][8] = {
    {1.f, 0.f, 0.1f, 0.f,   0.f,  0.f,  0.f,  0.f },
    {0.f, 1.f, 0.f,  0.1f,  0.f,  0.f,  0.f,  0.f },
    {0.f, 0.f, 0.75f,0.f,   0.5f,-0.5f, 0.f,  0.f },
    {0.f, 0.f, 0.f,  0.75f, 0.f,  0.f,  0.5f,-0.5f}};
__device__ __constant__ float c_q[8] = {1.f,1.f,0.1f,0.1f, 0.1f,0.1f,0.1f,0.1f};

// ---------------------------------------------------------------------------
// Setup kernel: one wave32. Riccati recursion for the batch-invariant part.
// Per t:  Q = diag(q) + F' V F   (two WMMA f32 16x16x4 ops)
//         G = -inv(Quu); K = G*Qux; V <- Qxx + Qxu*K
// Writes per-t constants to cst: [K(16) | G(16) | Qxu(16)] * 5  (240 floats).
// ---------------------------------------------------------------------------
__global__ __launch_bounds__(32) void mpc_riccati_setup(float* __restrict__ cst)
{
    const int lane = threadIdx.x;
    const int half = lane >> 4;   // 0 -> K{0,1}, 1 -> K{2,3}
    const int l16  = lane & 15;

    __shared__ float sV[16];   // 4x4 cost-to-go (row major)
    __shared__ float sW[32];   // 4x8  W = V*F
    __shared__ float sQ[64];   // 8x8  Q

    if (lane < 16) sV[lane] = 0.f;
    __syncthreads();

    for (int t = T_HORIZON - 1; t >= 0; --t) {
        // ---- WMMA #1:  W(4x8) = V(4x4) x F(4x8), padded into 16x16x4 ----
        v2f a1, b1;
        // A operand (16x4): rows 0-3 = V; lane l16 = row, half selects K pair
        if (l16 < 4) { a1.x = sV[l16*4 + 2*half + 0]; a1.y = sV[l16*4 + 2*half + 1]; }
        else         { a1.x = 0.f; a1.y = 0.f; }
        // B operand (4x16): col = l16 (cols 0-7 = F), rows per K pair
        b1.x = (l16 < 8) ? c_F[2*half + 0][l16] : 0.f;
        b1.y = (l16 < 8) ? c_F[2*half + 1][l16] : 0.f;
        v8f d1 = {};
        d1 = __builtin_amdgcn_wmma_f32_16x16x4_f32(false, a1, false, b1,
                                                   (short)0, d1, false, false);
        // D layout: VGPR r = row r (lanes 0-15), col = lane. Keep rows 0-3, cols 0-7.
        if (half == 0 && l16 < 8) {
            sW[0*8 + l16] = d1[0]; sW[1*8 + l16] = d1[1];
            sW[2*8 + l16] = d1[2]; sW[3*8 + l16] = d1[3];
        }
        __syncthreads();

        // ---- WMMA #2:  M(8x8) = F'(8x4) x W(4x8) ----
        v2f a2, b2;
        if (l16 < 8) { a2.x = c_F[2*half + 0][l16]; a2.y = c_F[2*half + 1][l16]; }
        else         { a2.x = 0.f; a2.y = 0.f; }
        b2.x = (l16 < 8) ? sW[(2*half + 0)*8 + l16] : 0.f;
        b2.y = (l16 < 8) ? sW[(2*half + 1)*8 + l16] : 0.f;
        v8f d2 = {};
        d2 = __builtin_amdgcn_wmma_f32_16x16x4_f32(false, a2, false, b2,
                                                   (short)0, d2, false, false);
        // Q = diag(q) + M  (rows 0-7, cols 0-7)
        if (half == 0 && l16 < 8) {
            #pragma unroll
            for (int r = 0; r < 8; ++r)
                sQ[r*8 + l16] = d2[r] + ((r == l16) ? c_q[r] : 0.f);
        }
        __syncthreads();

        // ---- scalar tail: 4x4 SPD inverse, gains, V update ----
        if (lane == 0) {
            float Qxu[4][4], G[4][4], K[4][4];
            float M4[4][8];
            #pragma unroll
            for (int i = 0; i < 4; ++i) {
                #pragma unroll
                for (int j = 0; j < 4; ++j) {
                    Qxu[i][j] = sQ[i*8 + (4 + j)];
                    M4[i][j]  = sQ[(4 + i)*8 + (4 + j)];   // Quu
                    M4[i][4+j] = (i == j) ? 1.f : 0.f;
                }
            }
            // Gauss-Jordan (Quu is SPD with diag >= 0.1: no pivoting needed)
            #pragma unroll
            for (int col = 0; col < 4; ++col) {
                float pv = 1.f / M4[col][col];
                #pragma unroll
                for (int j = 0; j < 8; ++j) M4[col][j] *= pv;
                #pragma unroll
                for (int r = 0; r < 4; ++r) if (r != col) {
                    float f = M4[r][col];
                    #pragma unroll
                    for (int j = 0; j < 8; ++j) M4[r][j] -= f * M4[col][j];
                }
            }
            #pragma unroll
            for (int i = 0; i < 4; ++i)
                #pragma unroll
                for (int j = 0; j < 4; ++j) G[i][j] = -M4[i][4 + j];
            // K = G * Qux
            #pragma unroll
            for (int u = 0; u < 4; ++u)
                #pragma unroll
                for (int x = 0; x < 4; ++x) {
                    float s = 0.f;
                    #pragma unroll
                    for (int m = 0; m < 4; ++m) s += G[u][m] * sQ[(4 + m)*8 + x];
                    K[u][x] = s;
                }
            // V <- Qxx + Qxu*K   (== Qxx - Qxu Quu^-1 Qux)
            float Vn[16];
            #pragma unroll
            for (int i = 0; i < 4; ++i)
                #pragma unroll
                for (int j = 0; j < 4; ++j) {
                    float s = sQ[i*8 + j];
                    #pragma unroll
                    for (int u = 0; u < 4; ++u) s += Qxu[i][u] * K[u][j];
                    Vn[i*4 + j] = s;
                }
            float* o = cst + t * 48;
            #pragma unroll
            for (int i = 0; i < 16; ++i) o[i]      = (&K[0][0])[i];
            #pragma unroll
            for (int i = 0; i < 16; ++i) o[16 + i] = (&G[0][0])[i];
            #pragma unroll
            for (int i = 0; i < 16; ++i) o[32 + i] = (&Qxu[0][0])[i];
            #pragma unroll
            for (int i = 0; i < 16; ++i) sV[i] = Vn[i];
        }
        __syncthreads();
    }
}

// ---------------------------------------------------------------------------
// Main kernel: one thread per batch element, everything in registers.
// Constants broadcast from LDS (uniform-address ds_load_b128).
// ---------------------------------------------------------------------------
__global__ __launch_bounds__(256) void mpc_solve(const float* __restrict__ x_init,
                                                 const float* __restrict__ xd,
                                                 const float* __restrict__ cst,
                                                 float* __restrict__ out, int B)
{
    // per t: rows 0-3 = K, 4-7 = G (= -Quu^-1), 8-11 = Qxu   (float4 rows)
    __shared__ float4 sC[T_HORIZON * 12];
    if (threadIdx.x < T_HORIZON * 12)
        sC[threadIdx.x] = ((const float4*)cst)[threadIdx.x];
    __syncthreads();

    const int i = blockIdx.x * blockDim.x + threadIdx.x;
    if (i >= B) return;

    const float4 x0  = ((const float4*)x_init)[i];
    const float4 xdv = ((const float4*)xd)[i];
    const float cx0 = -xdv.x, cx1 = -xdv.y;
    const float cx2 = -0.31622776601683794f * xdv.z;
    const float cx3 = -0.31622776601683794f * xdv.w;

    float u[T_HORIZON][4];
    #pragma unroll
    for (int t = 0; t < T_HORIZON; ++t)
        { u[t][0] = 0.f; u[t][1] = 0.f; u[t][2] = 0.f; u[t][3] = 0.f; }

    float chx[T_HORIZON][4], chu[T_HORIZON][4], kk[T_HORIZON][4];

    for (int it = 0; it < 100; ++it) {
        // ---- rollout + c_hat = c + diag(q) * [x;u] ----
        float s0 = x0.x, s1 = x0.y, s2 = x0.z, s3 = x0.w;
        #pragma unroll
        for (int t = 0; t < T_HORIZON; ++t) {
            chx[t][0] = cx0 + s0;          chx[t][1] = cx1 + s1;
            chx[t][2] = cx2 + 0.1f * s2;   chx[t][3] = cx3 + 0.1f * s3;
            chu[t][0] = 0.01f + 0.1f * u[t][0];
            chu[t][1] = 0.01f + 0.1f * u[t][1];
            chu[t][2] = 0.01f + 0.1f * u[t][2];
            chu[t][3] = 0.01f + 0.1f * u[t][3];
            const float n0 = s0 + 0.1f * s2;
            const float n1 = s1 + 0.1f * s3;
            const float n2 = 0.75f * s2 + 0.5f * (u[t][0] - u[t][1]);
            const float n3 = 0.75f * s3 + 0.5f * (u[t][2] - u[t][3]);
            s0 = n0; s1 = n1; s2 = n2; s3 = n3;
        }
        // ---- backward: linear terms only (quadratic part precomputed) ----
        float v0 = 0.f, v1 = 0.f, v2 = 0.f, v3 = 0.f;
        #pragma unroll
        for (int t = T_HORIZON - 1; t >= 0; --t) {
            const float qu0 = chu[t][0] + 0.5f * v2;
            const float qu1 = chu[t][1] - 0.5f * v2;
            const float qu2 = chu[t][2] + 0.5f * v3;
            const float qu3 = chu[t][3] - 0.5f * v3;
            const float4* G = &sC[t*12 + 4];
            const float k0 = G[0].x*qu0 + G[0].y*qu1 + G[0].z*qu2 + G[0].w*qu3;
            const float k1 = G[1].x*qu0 + G[1].y*qu1 + G[1].z*qu2 + G[1].w*qu3;
            const float k2 = G[2].x*qu0 + G[2].y*qu1 + G[2].z*qu2 + G[2].w*qu3;
            const float k3 = G[3].x*qu0 + G[3].y*qu1 + G[3].z*qu2 + G[3].w*qu3;
            kk[t][0] = k0; kk[t][1] = k1; kk[t][2] = k2; kk[t][3] = k3;
            const float qx0 = chx[t][0] + v0;
            const float qx1 = chx[t][1] + v1;
            const float qx2 = chx[t][2] + 0.1f * v0 + 0.75f * v2;
            const float qx3 = chx[t][3] + 0.1f * v1 + 0.75f * v3;
            const float4* X = &sC[t*12 + 8];
            v0 = qx0 + X[0].x*k0 + X[0].y*k1 + X[0].z*k2 + X[0].w*k3;
            v1 = qx1 + X[1].x*k0 + X[1].y*k1 + X[1].z*k2 + X[1].w*k3;
            v2 = qx2 + X[2].x*k0 + X[2].y*k1 + X[2].z*k2 + X[2].w*k3;
            v3 = qx3 + X[3].x*k0 + X[3].y*k1 + X[3].z*k2 + X[3].w*k3;
        }
        // ---- clipped forward sweep ----
        float d0 = 0.f, d1 = 0.f, d2 = 0.f, d3 = 0.f;
        float delta = 0.f;
        #pragma unroll
        for (int t = 0; t < T_HORIZON; ++t) {
            const float4* K = &sC[t*12 + 0];
            const float du0 = kk[t][0] + K[0].x*d0 + K[0].y*d1 + K[0].z*d2 + K[0].w*d3;
            const float du1 = kk[t][1] + K[1].x*d0 + K[1].y*d1 + K[1].z*d2 + K[1].w*d3;
            const float du2 = kk[t][2] + K[2].x*d0 + K[2].y*d1 + K[2].z*d2 + K[2].w*d3;
            const float du3 = kk[t][3] + K[3].x*d0 + K[3].y*d1 + K[3].z*d2 + K[3].w*d3;
            const float un0 = fminf(fmaxf(u[t][0] + du0, 0.f), 1.f);
            const float un1 = fminf(fmaxf(u[t][1] + du1, 0.f), 1.f);
            const float un2 = fminf(fmaxf(u[t][2] + du2, 0.f), 1.f);
            const float un3 = fminf(fmaxf(u[t][3] + du3, 0.f), 1.f);
            const float e0 = un0 - u[t][0], e1 = un1 - u[t][1];
            const float e2 = un2 - u[t][2], e3 = un3 - u[t][3];
            delta = fmaxf(delta, fmaxf(fmaxf(fabsf(e0), fabsf(e1)),
                                       fmaxf(fabsf(e2), fabsf(e3))));
            const float nd0 = d0 + 0.1f * d2;
            const float nd1 = d1 + 0.1f * d3;
            const float nd2 = 0.75f * d2 + 0.5f * (e0 - e1);
            const float nd3 = 0.75f * d3 + 0.5f * (e2 - e3);
            d0 = nd0; d1 = nd1; d2 = nd2; d3 = nd3;
            u[t][0] = un0; u[t][1] = un1; u[t][2] = un2; u[t][3] = un3;
        }
        if (delta <= 0.01f) break;   // reference stop tolerance (EPS)
    }

    float4 r; r.x = u[0][0]; r.y = u[0][1]; r.z = u[0][2]; r.w = u[0][3];
    ((float4*)out)[i] = r;
}

extern "C" void kernel_launch(void* const* d_in, const int* in_sizes, int n_in,
                              void* d_out, int out_size, void* d_ws, size_t ws_size,
                              hipStream_t stream)
{
    const float* x_init = (const float*)d_in[0];
    const float* xd     = (const float*)d_in[1];
    float* cst = (float*)d_ws;                 // 240 floats of Riccati constants
    const int B = in_sizes[0] / 4;             // NS = 4

    mpc_riccati_setup<<<1, 32, 0, stream>>>(cst);
    mpc_solve<<<(B + 255) / 256, 256, 0, stream>>>(x_init, xd, cst,
                                                   (float*)d_out, B);
}